// Fine_Change_Moment2_28656021799221
// MI455X (gfx1250) — compile-verified
//
#include <hip/hip_runtime.h>

// MI455X / gfx1250, wave32.
// Two-phase memory-bound pipeline:
//   K1: cam[n,t] via V_WMMA_F32_16X16X4_F32 (A = 16 t-rows x 4 patch taps,
//       B = W[:,0] broadcast across 16 cols) + min/max-over-t normalize -> ws
//   K2: align_corners bilinear 128->256 upsample + final [B*S*S,T] layout -> out

typedef __attribute__((ext_vector_type(2))) float v2f;
typedef __attribute__((ext_vector_type(8))) float v8f;

#define Bb 8
#define Ss 256
#define Tt 64
#define Hh 128
// N pooled pixels = Bb*Hh*Hh = 131072 ; output pixels = Bb*Ss*Ss = 524288

__global__ __launch_bounds__(256) void cam_wmma_kernel(const float* __restrict__ inp,
                                                       const float* __restrict__ W,
                                                       float* __restrict__ cam) {
  const int lane = threadIdx.x & 31;
  const int wave = threadIdx.x >> 5;
  const int n = blockIdx.x * 8 + wave;        // one wave per pooled pixel n
  const int b  = n >> 14;                     // n = ((b*128)+h0)*128 + h1
  const int h0 = (n >> 7) & 127;
  const int h1 = n & 127;

  const bool hi = (lane >= 16);
  const int  Lh = lane & 15;

  // input[b, 2h0(+1), 2h1(+1), t]; taps p0/p1 at row 2h0, p2/p3 at row 2h0+1
  const int base00 = ((b * Ss + 2 * h0) * Ss + 2 * h1) * Tt;
  const float* rowbase = inp + base00 + (hi ? (Ss * Tt) : 0);

  // B matrix 4x16, all columns = W[:,0].
  // Layout: VGPR0 = row K0 (lanes 0-15) / K2 (lanes 16-31); VGPR1 = K1 / K3.
  v2f bm;
  bm[0] = hi ? W[2 * 6] : W[0 * 6];
  bm[1] = hi ? W[3 * 6] : W[1 * 6];

  const v8f zero = {};
  v8f d0, d1, d2, d3;
  // A 16x4 f32: lanes 0-15 -> (K0,K1) = taps (p0,p1) at t = tb+Lh,
  //             lanes 16-31 -> (K2,K3) = taps (p2,p3) at t = tb+Lh.
  {
    v2f a;
    a[0] = rowbase[ 0 + Lh]; a[1] = rowbase[Tt +  0 + Lh];
    d0 = __builtin_amdgcn_wmma_f32_16x16x4_f32(false, a, false, bm, (short)0, zero, false, false);
    a[0] = rowbase[16 + Lh]; a[1] = rowbase[Tt + 16 + Lh];
    d1 = __builtin_amdgcn_wmma_f32_16x16x4_f32(false, a, false, bm, (short)0, zero, false, false);
    a[0] = rowbase[32 + Lh]; a[1] = rowbase[Tt + 32 + Lh];
    d2 = __builtin_amdgcn_wmma_f32_16x16x4_f32(false, a, false, bm, (short)0, zero, false, false);
    a[0] = rowbase[48 + Lh]; a[1] = rowbase[Tt + 48 + Lh];
    d3 = __builtin_amdgcn_wmma_f32_16x16x4_f32(false, a, false, bm, (short)0, zero, false, false);
  }

  // Lane (half) holds cam[t] for t = 16k + v (+8 for hi half), k in d0..d3, v = elem.
  // min/max over this lane's 32 values, then merge halves via shfl_xor(16).
  float mn, mx;
  {
    v8f mnv, mxv;
#pragma unroll
    for (int e = 0; e < 8; ++e) {
      mnv[e] = fminf(fminf(d0[e], d1[e]), fminf(d2[e], d3[e]));
      mxv[e] = fmaxf(fmaxf(d0[e], d1[e]), fmaxf(d2[e], d3[e]));
    }
    mn = mnv[0]; mx = mxv[0];
#pragma unroll
    for (int e = 1; e < 8; ++e) { mn = fminf(mn, mnv[e]); mx = fmaxf(mx, mxv[e]); }
  }
  mn = fminf(mn, __shfl_xor(mn, 16, 32));
  mx = fmaxf(mx, __shfl_xor(mx, 16, 32));
  const float inv = 1.0f / (mx - mn);

  // Each lane extracts 2 distinct t values (static select tree on lane bits):
  //   k  = (lane>>2)&3  selects d0..d3
  //   pi = lane&3       selects element pair (2pi, 2pi+1)
  //   t  = 16k + 2pi (+8 for hi half)
  v8f s0 = (lane & 8) ? d2 : d0;
  v8f s1 = (lane & 8) ? d3 : d1;
  v8f dk = (lane & 4) ? s1 : s0;
  float e0 = (lane & 1) ? dk[2] : dk[0];
  float e1 = (lane & 1) ? dk[3] : dk[1];
  float e2 = (lane & 1) ? dk[6] : dk[4];
  float e3 = (lane & 1) ? dk[7] : dk[5];
  float xa = (lane & 2) ? e2 : e0;
  float xb = (lane & 2) ? e3 : e1;

  const int tt = ((lane >> 2) & 3) * 16 + (lane & 3) * 2 + (hi ? 8 : 0);
  float2 o;
  o.x = (xa - mn) * inv;
  o.y = (xb - mn) * inv;
  *(float2*)(cam + ((size_t)n << 6) + tt) = o;
}

__global__ __launch_bounds__(256) void upsample_kernel(const float* __restrict__ cam,
                                                       float* __restrict__ out) {
  const int lane = threadIdx.x & 31;
  const int pix = blockIdx.x * 8 + (threadIdx.x >> 5);   // one wave per output pixel
  const int b = pix >> 16;
  const int y = (pix >> 8) & 255;
  const int x = pix & 255;

  const float r = 127.0f / 255.0f;                       // (H-1)/(S-1), align_corners
  float fy = (float)y * r; int y0 = (int)fy; float wy = fy - (float)y0;
  int y1 = (y0 < Hh - 1) ? y0 + 1 : Hh - 1;
  float fx = (float)x * r; int x0 = (int)fx; float wx = fx - (float)x0;
  int x1 = (x0 < Hh - 1) ? x0 + 1 : Hh - 1;

  const int t = lane << 1;
  const size_t pb = (size_t)b << 14;                     // b * 128*128
  const float* c00 = cam + (((pb + (size_t)y0 * Hh + x0) << 6) + t);
  const float* c01 = cam + (((pb + (size_t)y0 * Hh + x1) << 6) + t);
  const float* c10 = cam + (((pb + (size_t)y1 * Hh + x0) << 6) + t);
  const float* c11 = cam + (((pb + (size_t)y1 * Hh + x1) << 6) + t);
  float2 v00 = *(const float2*)c00;
  float2 v01 = *(const float2*)c01;
  float2 v10 = *(const float2*)c10;
  float2 v11 = *(const float2*)c11;

  // Match reference op order: interpolate along H first, then W.
  float h0x = v00.x * (1.0f - wy) + v10.x * wy;
  float h1x = v01.x * (1.0f - wy) + v11.x * wy;
  float h0y = v00.y * (1.0f - wy) + v10.y * wy;
  float h1y = v01.y * (1.0f - wy) + v11.y * wy;
  float2 o;
  o.x = h0x * (1.0f - wx) + h1x * wx;
  o.y = h0y * (1.0f - wx) + h1y * wx;

  *(float2*)(out + (((size_t)pix) << 6) + t) = o;
}

extern "C" void kernel_launch(void* const* d_in, const int* in_sizes, int n_in,
                              void* d_out, int out_size, void* d_ws, size_t ws_size,
                              hipStream_t stream) {
  const float* inp = (const float*)d_in[0];   // [8,256,256,64] f32
  const float* W   = (const float*)d_in[1];   // [4,6] f32 (only column 0 is live)
  float* out = (float*)d_out;                 // [8*256*256, 64] f32
  float* cam = (float*)d_ws;                  // needs 8*128*128*64*4 = 32 MiB scratch

  // 131072 pooled pixels, 8 per 256-thread block (1 wave / pixel)
  cam_wmma_kernel<<<16384, 256, 0, stream>>>(inp, W, cam);
  // 524288 output pixels, 8 per 256-thread block (1 wave / pixel)
  upsample_kernel<<<65536, 256, 0, stream>>>(cam, out);
}